// BinsChamferLoss_47150150976039
// MI455X (gfx1250) — compile-verified
//
#include <hip/hip_runtime.h>

// ---------------------------------------------------------------------------
// BinsChamferLoss on MI455X (gfx1250, wave32)
//
// B=8, P=256 bins, M=240*320=76800 pixels. Total data 2.4MB (<< 192MB L2),
// ~157M |c-t| min-evals -> VALU bound, a few microseconds. Brute-force P x M
// with targets streamed Global -> LDS via CDNA5 async-copy
// (global_load_async_to_lds_b128 + s_wait_asynccnt, ASYNCcnt-tracked) with a
// double buffer; per-(b,p) float-min via uint atomicMin (distances >= 0, so
// f32 bit patterns order monotonically).
//
// Grid is 2D (tiles x batches) to avoid an integer division in the prologue.
// ---------------------------------------------------------------------------

#define NBINS_P   256
#define TILE_M    3072              // targets per block (25 tiles per batch)
#define CHUNK     1024              // floats per LDS chunk = 256 lanes * 4
#define NCHUNK    (TILE_M / CHUNK)  // 3 chunks -> real double-buffer overlap
#define MIN_DEPTH 0.001f

#if defined(__HIP_DEVICE_COMPILE__) &&                                       \
    __has_builtin(__builtin_amdgcn_global_load_async_to_lds_b128) &&         \
    __has_builtin(__builtin_amdgcn_s_wait_asynccnt)
#define USE_ASYNC 1
#else
#define USE_ASYNC 0
#endif

typedef int v4i __attribute__((ext_vector_type(4)));
typedef __attribute__((address_space(1))) v4i* gv4p;  // global int4*
typedef __attribute__((address_space(3))) v4i* lv4p;  // LDS int4*

// Copy CHUNK floats (gsrc[0..1023]) into LDS ldst, one b128 per lane.
__device__ __forceinline__ void lds_copy_chunk(const float* gsrc, float* ldst,
                                               int tid) {
#if USE_ASYNC
  __builtin_amdgcn_global_load_async_to_lds_b128(
      (gv4p)(void*)(gsrc + tid * 4),
      (lv4p)(void*)(ldst + tid * 4),
      /*offset=*/0, /*cpol=*/0);
#else
  ((float4*)ldst)[tid] = ((const float4*)gsrc)[tid];
#endif
}

__device__ __forceinline__ void async_wait_le1() {
#if USE_ASYNC
  __builtin_amdgcn_s_wait_asynccnt(1);  // oldest async load done (in-order)
#endif
}
__device__ __forceinline__ void async_wait_le0() {
#if USE_ASYNC
  __builtin_amdgcn_s_wait_asynccnt(0);
#endif
}

__global__ __launch_bounds__(256) void chamfer_min_kernel(
    const float* __restrict__ bins, const float* __restrict__ tgt,
    unsigned int* __restrict__ ws, int M) {
  __shared__ float sbuf[2][CHUNK];  // 8KB double buffer

  const int tid  = threadIdx.x;   // one bin center per thread
  const int tile = blockIdx.x;    // which TILE_M slice of this batch
  const int b    = blockIdx.y;    // batch index

  const float* binb = bins + b * (NBINS_P + 1);
  const float  c    = 0.5f * (binb[tid] + binb[tid + 1]);

  const float* gbase = tgt + (size_t)b * (size_t)M + (size_t)tile * TILE_M;

  // Prologue: kick off chunk 0 (async DMA into LDS, ASYNCcnt-tracked).
  lds_copy_chunk(gbase, sbuf[0], tid);

  float d = __builtin_inff();
  for (int k = 0; k < NCHUNK; ++k) {
    if (k + 1 < NCHUNK) {
      lds_copy_chunk(gbase + (k + 1) * CHUNK, sbuf[(k + 1) & 1], tid);
      async_wait_le1();  // chunk k landed for this wave (loads retire in order)
    } else {
      async_wait_le0();
    }
    __syncthreads();  // all waves' chunk-k data visible in LDS

    const float4* v4 = (const float4*)sbuf[k & 1];
#pragma unroll 8
    for (int i = 0; i < CHUNK / 4; ++i) {
      float4 v = v4[i];  // broadcast read: all lanes same addr, no conflicts
      float t0 = (v.x >= MIN_DEPTH) ? v.x : 0.0f;
      float t1 = (v.y >= MIN_DEPTH) ? v.y : 0.0f;
      float t2 = (v.z >= MIN_DEPTH) ? v.z : 0.0f;
      float t3 = (v.w >= MIN_DEPTH) ? v.w : 0.0f;
      d = fminf(d, __builtin_fabsf(c - t0));
      d = fminf(d, __builtin_fabsf(c - t1));
      d = fminf(d, __builtin_fabsf(c - t2));
      d = fminf(d, __builtin_fabsf(c - t3));
    }
    __syncthreads();  // everyone done reading before buffer is reused
  }

  // Non-negative floats order like their uint bit patterns.
  atomicMin(&ws[b * NBINS_P + tid], __float_as_uint(d));
}

__global__ void init_ws_kernel(unsigned int* ws, int n) {
  int i = blockIdx.x * blockDim.x + threadIdx.x;
  if (i < n) ws[i] = 0x7F800000u;  // +inf bit pattern
}

__global__ __launch_bounds__(256) void reduce_sum_kernel(
    const unsigned int* __restrict__ ws, float* __restrict__ out, int n) {
  __shared__ float s[256];
  float acc = 0.0f;
  for (int i = threadIdx.x; i < n; i += 256) acc += __uint_as_float(ws[i]);
  s[threadIdx.x] = acc;
  __syncthreads();
  for (int off = 128; off > 0; off >>= 1) {
    if ((int)threadIdx.x < off) s[threadIdx.x] += s[threadIdx.x + off];
    __syncthreads();
  }
  if (threadIdx.x == 0) out[0] = s[0];
}

extern "C" void kernel_launch(void* const* d_in, const int* in_sizes, int n_in,
                              void* d_out, int out_size, void* d_ws,
                              size_t ws_size, hipStream_t stream) {
  const float* bins = (const float*)d_in[0];          // [B, P+1]
  const float* tgt  = (const float*)d_in[1];          // [B, 1, H, W]

  const int B     = in_sizes[0] / (NBINS_P + 1);      // 8
  const int M     = in_sizes[1] / B;                  // 76800
  const int tiles = M / TILE_M;                       // 25

  unsigned int* ws  = (unsigned int*)d_ws;            // B*P = 2048 slots
  float*        out = (float*)d_out;
  const int NP = B * NBINS_P;

  init_ws_kernel<<<(NP + 255) / 256, 256, 0, stream>>>(ws, NP);
  chamfer_min_kernel<<<dim3(tiles, B), 256, 0, stream>>>(bins, tgt, ws, M);
  reduce_sum_kernel<<<1, 256, 0, stream>>>(ws, out, NP);
}